// ElevationBoundaryContinuityLoss_25975962206364
// MI455X (gfx1250) — compile-verified
//
#include <hip/hip_runtime.h>
#include <hip/hip_bf16.h>
#include <math.h>

// ---------------------------------------------------------------------------
// ElevationBoundaryContinuityLoss for MI455X (gfx1250, wave32)
//   - fused stencil losses: one pass over L2-resident inputs (bandwidth bound)
//   - 2D FFT via four-step DFT (1024 = 32x32) on V_WMMA_F32_16X16X4_F32
//   - LDS feeding via TDM (tensor_load_to_lds) and async global->LDS ops
// ---------------------------------------------------------------------------

#define HH 1024
#define WW 1024
#define NB 8
#define HWSZ (HH * WW)

#define TWO_PI_F 6.283185307179586f

typedef float v2f __attribute__((ext_vector_type(2)));
typedef float v8f __attribute__((ext_vector_type(8)));
typedef unsigned int u32x4 __attribute__((ext_vector_type(4)));
typedef int i32x4 __attribute__((ext_vector_type(4)));
typedef int i32x8 __attribute__((ext_vector_type(8)));
typedef int gv2i __attribute__((vector_size(8)));  // matches async-LDS builtin

#if defined(__gfx1250__) && __has_builtin(__builtin_amdgcn_tensor_load_to_lds)
#define HAVE_TDM 1
#else
#define HAVE_TDM 0
#endif
#if defined(__gfx1250__) && \
    __has_builtin(__builtin_amdgcn_global_load_async_to_lds_b64)
#define HAVE_ASYNC_LDS 1
#else
#define HAVE_ASYNC_LDS 0
#endif

__device__ __forceinline__ void wait_asynccnt0() {
#if defined(__gfx1250__)
#if __has_builtin(__builtin_amdgcn_s_wait_asynccnt)
  __builtin_amdgcn_s_wait_asynccnt(0);
#else
  asm volatile("s_wait_asynccnt 0x0" ::: "memory");
#endif
#endif
}

// ---------------------------------------------------------------------------
// WMMA fragment helpers (wave32, f32 16x16x4), layouts per CDNA5 ISA §7.12.2
// ---------------------------------------------------------------------------
// A: 16x4 f32.  lane L: row = L&15 ; holds K = {0,1} (L<16) or {2,3} (L>=16)
__device__ __forceinline__ v2f ldA(const float* M, int m0, int k0, int lane) {
  int row = m0 + (lane & 15);
  int kb  = k0 + ((lane >> 4) << 1);
  v2f a;
  a.x = M[row * 32 + kb];
  a.y = M[row * 32 + kb + 1];
  return a;
}
// B: 4x16 f32.  lane L: col = L&15 ; holds K = {0,1} (L<16) or {2,3} (L>=16)
__device__ __forceinline__ v2f ldB(const float* M, int k0, int n0, int lane) {
  int col = n0 + (lane & 15);
  int kb  = k0 + ((lane >> 4) << 1);
  v2f b;
  b.x = M[kb * 32 + col];
  b.y = M[(kb + 1) * 32 + col];
  return b;
}

// acc += asign * A(32x32 tile rows m0..m0+15, K=0..31) * B(K, cols n0..n0+15)
__device__ __forceinline__ v8f mm16x16k32(const float* A, const float* B,
                                          int m0, int n0, int lane,
                                          v8f acc, float asign) {
#pragma unroll
  for (int kk = 0; kk < 8; ++kk) {
    v2f a = ldA(A, m0, kk * 4, lane);
    a.x *= asign;
    a.y *= asign;
    v2f b = ldB(B, kk * 4, n0, lane);
    acc = __builtin_amdgcn_wmma_f32_16x16x4_f32(
        /*neg_a=*/false, a, /*neg_b=*/false, b,
        /*c_mod=*/(short)0, acc, /*reuse_a=*/false, /*reuse_b=*/false);
  }
  return acc;
}

// ---------------------------------------------------------------------------
// 1024-point DFT as four-step radix-32:  X = transpose_pack( (D32*M .* T) * D32 )
//   pass1 (inR != null): transform t = image row, contiguous read  (TDM load)
//   pass2 (inR == null): transform t = column of row-spectrum, strided read
//                        (per-lane async global->LDS gather)
// out[t*1024 + n] contiguous; 2D spectrum ends up transposed (harmless: the
// hf mask and the scalar means are transpose-invariant).
// 128 threads = 4 waves; wave w owns 16x16 output tile (w>>1, w&1).
// ---------------------------------------------------------------------------
__global__ void __launch_bounds__(128)
fft1024_kernel(const float* __restrict__ inR, const float2* __restrict__ inC,
               float2* __restrict__ out) {
  __shared__ float Dr[1024], Di[1024];  // DFT-32 matrix (re, im)
  __shared__ float Tr[1024], Ti[1024];  // twiddles W_1024^{k2*n1}
  __shared__ float Xr[1024], Xi[1024];  // input 32x32 / reused for result
  __shared__ float Gr[1024], Gi[1024];  // twiddled stage-1 result
  __shared__ float2 XC[1024];           // AoS staging for async gather (pass2)

  const int tid = threadIdx.x;
  const int t = blockIdx.x;
  const bool is_real = (inR != nullptr);

  // ---- stage the input slab into LDS on the CDNA5 async paths -------------
  if (is_real) {
#if HAVE_TDM
    if (tid < 32) {  // TDM ignores EXEC; issue the DMA from one wave only
      unsigned long long ga =
          (unsigned long long)(const void*)(inR + (size_t)t * 1024);
      unsigned int laddr = (unsigned int)(size_t)&Xr[0];  // LDS byte offset
      // D# group0: count=1, lds_addr, global_addr[56:0], type=2
      u32x4 g0 = {1u, laddr, (unsigned int)ga,
                  (unsigned int)((ga >> 32) & 0x01FFFFFFu) | (2u << 30)};
      // D# group1: data_size=4B; tensor 1024x1024 (stride 1024); tile 1024x1
      i32x8 g1 = {(int)0x00020000,      // data_size=2 (4 bytes)
                  (int)(1024u << 16),   // tensor_dim0[15:0]
                  (int)(1024u << 16),   // tensor_dim1[15:0]
                  (int)(1024u << 16),   // tile_dim0=1024
                  1,                    // tile_dim1=1
                  1024,                 // tensor_dim0_stride[31:0]
                  0, 0};
      i32x4 z4 = {0, 0, 0, 0};
#if __clang_major__ >= 23
      i32x8 z8 = {0, 0, 0, 0, 0, 0, 0, 0};
      __builtin_amdgcn_tensor_load_to_lds(g0, g1, z4, z4, z8, 0);
#else
      __builtin_amdgcn_tensor_load_to_lds(g0, g1, z4, z4, 0);
#endif
      __builtin_amdgcn_s_wait_tensorcnt(0);
    }
#else
    for (int i = tid; i < 1024; i += 128) Xr[i] = inR[t * 1024 + i];
#endif
    for (int i = tid; i < 1024; i += 128) Xi[i] = 0.0f;
  } else {
#if HAVE_ASYNC_LDS
    for (int i = tid; i < 1024; i += 128) {
      const float2* gp = inC + (size_t)i * 1024 + t;
      __builtin_amdgcn_global_load_async_to_lds_b64(
          (__attribute__((address_space(1))) gv2i*)(unsigned long long)(
              const void*)gp,
          (__attribute__((address_space(3))) gv2i*)(unsigned int)(size_t)&XC[i],
          0, 0);
    }
    wait_asynccnt0();
#else
    for (int i = tid; i < 1024; i += 128) {
      if (i + 128 < 1024) __builtin_prefetch(&inC[(i + 128) * 1024 + t], 0, 1);
      XC[i] = inC[(size_t)i * 1024 + t];
    }
#endif
  }

  // ---- DFT-32 matrix and 1024-pt twiddles ---------------------------------
  for (int i = tid; i < 1024; i += 128) {
    int r = i >> 5, c = i & 31;
    int m32 = (r * c) & 31;
    float a32 = TWO_PI_F * (float)m32 * (1.0f / 32.0f);
    Dr[i] = __cosf(a32);
    Di[i] = -__sinf(a32);
    int m1k = (r * c) & 1023;
    float a1k = TWO_PI_F * (float)m1k * (1.0f / 1024.0f);
    Tr[i] = __cosf(a1k);
    Ti[i] = -__sinf(a1k);
  }
  __syncthreads();
  if (!is_real) {  // split AoS staging into SoA operand matrices
    for (int i = tid; i < 1024; i += 128) {
      float2 v = XC[i];
      Xr[i] = v.x;
      Xi[i] = v.y;
    }
    __syncthreads();
  }

  const int wave = tid >> 5;
  const int lane = tid & 31;
  const int m0 = (wave >> 1) << 4;
  const int n0 = (wave & 1) << 4;

  // ---- stage 1: G = D32 * X (complex) -------------------------------------
  v8f gr = {0.f, 0.f, 0.f, 0.f, 0.f, 0.f, 0.f, 0.f};
  v8f gi = {0.f, 0.f, 0.f, 0.f, 0.f, 0.f, 0.f, 0.f};
  gr = mm16x16k32(Dr, Xr, m0, n0, lane, gr, 1.0f);
  gi = mm16x16k32(Di, Xr, m0, n0, lane, gi, 1.0f);
  if (!is_real) {
    gr = mm16x16k32(Di, Xi, m0, n0, lane, gr, -1.0f);
    gi = mm16x16k32(Dr, Xi, m0, n0, lane, gi, 1.0f);
  }

  // ---- twiddle G' = G .* T and park in LDS --------------------------------
#pragma unroll
  for (int r8 = 0; r8 < 8; ++r8) {
    int row = m0 + r8 + ((lane >> 4) << 3);
    int col = n0 + (lane & 15);
    int idx = row * 32 + col;
    float tr = Tr[idx], ti = Ti[idx];
    float grv = gr[r8], giv = gi[r8];
    Gr[idx] = grv * tr - giv * ti;
    Gi[idx] = grv * ti + giv * tr;
  }
  __syncthreads();

  // ---- stage 2: R = G' * D32 (complex) ------------------------------------
  v8f rr = {0.f, 0.f, 0.f, 0.f, 0.f, 0.f, 0.f, 0.f};
  v8f ri = {0.f, 0.f, 0.f, 0.f, 0.f, 0.f, 0.f, 0.f};
  rr = mm16x16k32(Gr, Dr, m0, n0, lane, rr, 1.0f);
  rr = mm16x16k32(Gi, Di, m0, n0, lane, rr, -1.0f);
  ri = mm16x16k32(Gr, Di, m0, n0, lane, ri, 1.0f);
  ri = mm16x16k32(Gi, Dr, m0, n0, lane, ri, 1.0f);

#pragma unroll
  for (int r8 = 0; r8 < 8; ++r8) {
    int row = m0 + r8 + ((lane >> 4) << 3);
    int col = n0 + (lane & 15);
    Xr[row * 32 + col] = rr[r8];
    Xi[row * 32 + col] = ri[r8];
  }
  __syncthreads();

  // X_full[k2 + 32*k1] = R[k2][k1] ; linear contiguous store
  for (int n = tid; n < 1024; n += 128) {
    int k2 = n & 31, k1 = n >> 5;
    out[t * 1024 + n] = make_float2(Xr[k2 * 32 + k1], Xi[k2 * 32 + k1]);
  }
}

// ---------------------------------------------------------------------------
// Fused stencil losses: grad (3 widths), smooth, slope. 16x16 output tiles.
//   acc[0..2]=w1 {slope,dir,curv}, acc[3..5]=w2, acc[6..8]=w3,
//   acc[9]=smooth, acc[10]=slope-cont, acc[11]=fft-mag, acc[12]=fft-phase
// ---------------------------------------------------------------------------
__device__ __forceinline__ void sobel3(const float* s, int ld, int c,
                                       float& gx, float& gy) {
  float a00 = s[c - ld - 1], a01 = s[c - ld], a02 = s[c - ld + 1];
  float a10 = s[c - 1], a12 = s[c + 1];
  float a20 = s[c + ld - 1], a21 = s[c + ld], a22 = s[c + ld + 1];
  gx = -a00 + a02 - 2.f * a10 + 2.f * a12 - a20 + a22;
  gy = -a00 - 2.f * a01 - a02 + a20 + 2.f * a21 + a22;
}

__global__ void __launch_bounds__(256)
stencil_loss_kernel(const float* __restrict__ pred,
                    const float* __restrict__ targ,
                    const float* __restrict__ mask,
                    float* __restrict__ acc) {
  __shared__ float sM[22 * 22];   // mask, halo 3
  __shared__ float sP[20 * 20];   // pred, halo 2
  __shared__ float sT[20 * 20];   // target, halo 2
  __shared__ float sPm[18 * 18];  // |grad pred|, halo 1
  __shared__ float sTm[18 * 18];  // |grad target|, halo 1
  __shared__ float red[256];

  const int tid = threadIdx.x;
  const int ox = blockIdx.x * 16, oy = blockIdx.y * 16;
  const long base = (long)blockIdx.z * HWSZ;

  for (int i = tid; i < 22 * 22; i += 256) {
    int ly = i / 22 - 3, lx = i % 22 - 3;
    int gy = oy + ly, gx = ox + lx;
    sM[i] = (gy >= 0 && gy < HH && gx >= 0 && gx < WW)
                ? mask[base + (long)gy * WW + gx] : 0.f;
  }
  for (int i = tid; i < 20 * 20; i += 256) {
    int ly = i / 20 - 2, lx = i % 20 - 2;
    int gy = oy + ly, gx = ox + lx;
    bool in = (gy >= 0 && gy < HH && gx >= 0 && gx < WW);
    long g = base + (long)gy * WW + gx;
    sP[i] = in ? pred[g] : 0.f;
    sT[i] = in ? targ[g] : 0.f;
  }
  __syncthreads();

  for (int i = tid; i < 18 * 18; i += 256) {
    int ly = i / 18 - 1, lx = i % 18 - 1;
    int c = (ly + 2) * 20 + (lx + 2);
    float gx, gy;
    sobel3(sP, 20, c, gx, gy);
    float pm = sqrtf(gx * gx + gy * gy + 1e-8f);
    sobel3(sT, 20, c, gx, gy);
    float tm = sqrtf(gx * gx + gy * gy + 1e-8f);
    int iy = oy + ly, ix = ox + lx;  // feature maps are zero-padded outside
    bool in = (iy >= 0 && iy < HH && ix >= 0 && ix < WW);
    sPm[i] = in ? pm : 0.f;
    sTm[i] = in ? tm : 0.f;
  }
  __syncthreads();

  const int lx = tid & 15, ly = tid >> 4;
  const int cP = (ly + 2) * 20 + (lx + 2);
  const int cM = (ly + 3) * 22 + (lx + 3);
  const int cG = (ly + 1) * 18 + (lx + 1);

  // gradients / direction / curvature diffs
  float pgx, pgy, tgx, tgy;
  sobel3(sP, 20, cP, pgx, pgy);
  sobel3(sT, 20, cP, tgx, tgy);
  float pmag = sPm[cG];
  float tmag = sTm[cG];
  float pdir = atan2f(pgy, pgx + 1e-8f);
  float tdir = atan2f(tgy, tgx + 1e-8f);
  float pcur = 4.f * sP[cP] - sP[cP - 20] - sP[cP + 20] - sP[cP - 1] - sP[cP + 1];
  float tcur = 4.f * sT[cP] - sT[cP - 20] - sT[cP + 20] - sT[cP - 1] - sT[cP + 1];
  float slope_d = fabsf(pmag - tmag);
  float dir_d = fabsf(pdir - tdir);
  dir_d = fminf(dir_d, TWO_PI_F - dir_d);
  float curv_d = fabsf(pcur - tcur);

  // boundary masks for widths 1..3 via nested box sums of the 0/1 mask
  float s1 = 0.f, s2 = 0.f, s3 = 0.f;
  for (int dy = -3; dy <= 3; ++dy)
    for (int dx = -3; dx <= 3; ++dx) {
      float v = sM[cM + dy * 22 + dx];
      int ady = dy < 0 ? -dy : dy, adx = dx < 0 ? -dx : dx;
      int m = ady > adx ? ady : adx;
      s3 += v;
      if (m <= 2) s2 += v;
      if (m <= 1) s1 += v;
    }
  float bm1 = (s1 > 0.5f && s1 < 8.5f) ? 1.f : 0.f;
  float bm2 = (s2 > 0.5f && s2 < 24.5f) ? 1.f : 0.f;
  float bm3 = (s3 > 0.5f && s3 < 48.5f) ? 1.f : 0.f;

  // smoothness: local std of pred over 3x3
  float sum9 = 0.f, sq9 = 0.f;
  for (int dy = -1; dy <= 1; ++dy)
    for (int dx = -1; dx <= 1; ++dx) {
      float v = sP[cP + dy * 20 + dx];
      sum9 += v;
      sq9 += v * v;
    }
  float lm = sum9 * (1.f / 9.f);
  float lv = fmaxf(sq9 * (1.f / 9.f) - lm * lm, 1e-8f);
  float smooth_c = sqrtf(lv) * bm1;

  // slope continuity: Sobel of gradient magnitudes
  float px2, py2, tx2, ty2;
  sobel3(sPm, 18, cG, px2, py2);
  sobel3(sTm, 18, cG, tx2, ty2);
  float pch = sqrtf(px2 * px2 + py2 * py2 + 1e-8f);
  float tch = sqrtf(tx2 * tx2 + ty2 * ty2 + 1e-8f);
  float slope_c = fabsf(pch - tch) * bm1;

  float vals[11];
  vals[0] = slope_d * bm1; vals[1] = dir_d * bm1; vals[2] = curv_d * bm1;
  vals[3] = slope_d * bm2; vals[4] = dir_d * bm2; vals[5] = curv_d * bm2;
  vals[6] = slope_d * bm3; vals[7] = dir_d * bm3; vals[8] = curv_d * bm3;
  vals[9] = smooth_c;      vals[10] = slope_c;

  for (int a = 0; a < 11; ++a) {
    red[tid] = vals[a];
    __syncthreads();
    for (int s = 128; s > 0; s >>= 1) {
      if (tid < s) red[tid] += red[tid + s];
      __syncthreads();
    }
    if (tid == 0) atomicAdd(&acc[a], red[0]);
    __syncthreads();
  }
}

// ---------------------------------------------------------------------------
// Spectrum compare: mag/phase MSE on the high-frequency radial mask
// ---------------------------------------------------------------------------
__global__ void __launch_bounds__(256)
spectrum_loss_kernel(const float2* __restrict__ sp, const float2* __restrict__ st,
                     float* __restrict__ acc) {
  __shared__ float redm[256];
  __shared__ float redp[256];
  const int tid = threadIdx.x;
  const int idx = blockIdx.x * 256 + tid;
  float vm = 0.f, vp = 0.f;
  if (idx < HWSZ) {
    int i = idx >> 10, j = idx & 1023;
    float di = (float)(i - 512), dj = (float)(j - 512);
    bool hf = (di * di + dj * dj) > 94371.84f;  // (0.3*1024)^2
    if (hf) {
      float2 P = sp[idx], T = st[idx];
      float pm = sqrtf(P.x * P.x + P.y * P.y);
      float tm = sqrtf(T.x * T.x + T.y * T.y);
      float dmg = pm - tm;
      vm = dmg * dmg;
      float d = atan2f(P.y, P.x) - atan2f(T.y, T.x);
      float w = atan2f(__sinf(d), __cosf(d));
      vp = w * w;
    }
  }
  redm[tid] = vm;
  redp[tid] = vp;
  __syncthreads();
  for (int s = 128; s > 0; s >>= 1) {
    if (tid < s) { redm[tid] += redm[tid + s]; redp[tid] += redp[tid + s]; }
    __syncthreads();
  }
  if (tid == 0) {
    atomicAdd(&acc[11], redm[0]);
    atomicAdd(&acc[12], redp[0]);
  }
}

__global__ void zero_acc_kernel(float* acc) {
  if (threadIdx.x < 16) acc[threadIdx.x] = 0.f;
}

__global__ void finalize_kernel(const float* __restrict__ acc,
                                float* __restrict__ out) {
  const float invN = 1.0f / (8.0f * 1024.0f * 1024.0f);
  float g = (1.0f * (acc[0] + acc[1]) + 2.0f * acc[2] +
             0.5f * (acc[3] + acc[4]) + 1.0f * acc[5] +
             0.5f * (acc[6] + acc[7]) + 1.0f * acc[8]) * invN / 3.0f;
  float f = (acc[11] + 2.0f * acc[12]) * invN;
  float sm = acc[9] * invN;
  float sl = acc[10] * invN;
  out[0] = 2.0f * g + 1.5f * f + 3.0f * sm + 2.0f * sl;
  out[1] = g;
  out[2] = f;
  out[3] = sm;
  out[4] = sl;
}

// ---------------------------------------------------------------------------
extern "C" void kernel_launch(void* const* d_in, const int* in_sizes, int n_in,
                              void* d_out, int out_size, void* d_ws, size_t ws_size,
                              hipStream_t stream) {
  const float* pred = (const float*)d_in[0];
  const float* targ = (const float*)d_in[1];
  const float* mask = (const float*)d_in[2];
  float* out = (float*)d_out;

  float* acc = (float*)d_ws;
  float2* bufR = (float2*)((char*)d_ws + 256);            // row-spectrum ping
  float2* bufP = bufR + (size_t)HWSZ;                     // pred 2D spectrum
  float2* bufT = bufP + (size_t)HWSZ;                     // target 2D spectrum

  zero_acc_kernel<<<1, 32, 0, stream>>>(acc);

  stencil_loss_kernel<<<dim3(WW / 16, HH / 16, NB), 256, 0, stream>>>(
      pred, targ, mask, acc);

  for (int b = 0; b < NB; ++b) {
    const float* pb = pred + (size_t)b * HWSZ;
    const float* tb = targ + (size_t)b * HWSZ;
    fft1024_kernel<<<HH, 128, 0, stream>>>(pb, nullptr, bufR);
    fft1024_kernel<<<WW, 128, 0, stream>>>(nullptr, bufR, bufP);
    fft1024_kernel<<<HH, 128, 0, stream>>>(tb, nullptr, bufR);
    fft1024_kernel<<<WW, 128, 0, stream>>>(nullptr, bufR, bufT);
    spectrum_loss_kernel<<<HWSZ / 256, 256, 0, stream>>>(bufP, bufT, acc);
  }

  finalize_kernel<<<1, 1, 0, stream>>>(acc, out);
}